// EquilibriumAgg_23398981829380
// MI455X (gfx1250) — compile-verified
//
#include <hip/hip_runtime.h>

// ---------------------------------------------------------------------------
// EquilibriumAgg on MI455X (gfx1250, wave32).  f32 WMMA 16x16x4 for all
// matmuls; one workgroup per graph runs all T=10 equilibrium steps fused.
// Branch-free tanh (v_exp_f32 + v_rcp_f32) to kill ocml divergence.
// ---------------------------------------------------------------------------

typedef __attribute__((ext_vector_type(2))) float v2f;
typedef __attribute__((ext_vector_type(8))) float v8f;

#define DIM    128   // in_dim
#define HD     64    // hidden
#define NODES  512
#define TSTEPS 10
#define ALPHA  0.1f
#define REGC   1e-3f
#define LNEPS  1e-5f
#define TILE   32    // nodes per LDS tile
#define LD1    132   // padded stride for 128-wide tiles (bank spread)
#define LD2    68    // padded stride for 64-wide tiles
#define LDP    20    // padded stride for 16-wide tile (p)
#define LDW3   17    // padded stride for W3pad [64][16]

static __device__ __forceinline__ v8f wmma4(v2f a, v2f b, v8f c) {
  // D = A(16x4,f32) x B(4x16,f32) + C(16x16,f32)
  return __builtin_amdgcn_wmma_f32_16x16x4_f32(false, a, false, b, (short)0, c,
                                               false, false);
}

// Branch-free tanh: sign(x) * (1 - 2/(exp(2|x|)+1)).
// __expf -> v_exp_f32 (saturates to +inf for large arg -> result +/-1),
// v_rcp_f32 for the divide; no exec-mask divergence, ~1ulp abs error.
static __device__ __forceinline__ float fast_tanh(float x) {
  float ax = __builtin_fabsf(x);
  float e  = __expf(ax + ax);
  float r  = __builtin_amdgcn_rcpf(e + 1.0f);
  float t  = __builtin_fmaf(-2.0f, r, 1.0f);
  return __builtin_copysignf(t, x);
}

// A fragment 16x4: lanes 0-15 hold K=k0,k0+1; lanes 16-31 hold K=k0+2,k0+3
static __device__ __forceinline__ v2f fragA(const float* base, int ld, int r0, int k0) {
  int l = threadIdx.x & 31;
  const float* p = base + (r0 + (l & 15)) * ld + k0 + ((l >> 4) << 1);
  v2f a; a.x = p[0]; a.y = p[1]; return a;
}
// B fragment 4x16 from row-major [K][N] matrix with stride ld
static __device__ __forceinline__ v2f fragB(const float* W, int ld, int k0, int n0) {
  int l = threadIdx.x & 31;
  const float* p = W + (k0 + ((l >> 4) << 1)) * ld + n0 + (l & 15);
  v2f b; b.x = p[0]; b.y = p[ld]; return b;
}
// B fragment where B = M^T, M row-major [N][K] stride ld (contiguous pair/lane)
static __device__ __forceinline__ v2f fragBT(const float* M, int ld, int k0, int n0) {
  int l = threadIdx.x & 31;
  const float* p = M + (n0 + (l & 15)) * ld + k0 + ((l >> 4) << 1);
  v2f b; b.x = p[0]; b.y = p[1]; return b;
}

// LayerNorm forward: 32 rows, 8 lanes/row, biased variance (torch default)
template <int C, int LD>
static __device__ __forceinline__ void ln_fwd(const float* a, float* h,
                                              const float* gm, const float* bt,
                                              float* stat) {
  int row = threadIdx.x >> 3, li = threadIdx.x & 7;
  const float* ar = a + row * LD;
  float s = 0.f, ss = 0.f;
#pragma unroll
  for (int c = li; c < C; c += 8) { float v = ar[c]; s += v; ss += v * v; }
  for (int m = 4; m; m >>= 1) { s += __shfl_xor(s, m, 8); ss += __shfl_xor(ss, m, 8); }
  float mean = s * (1.f / C);
  float var  = ss * (1.f / C) - mean * mean;
  float rstd = rsqrtf(var + LNEPS);
#pragma unroll
  for (int c = li; c < C; c += 8)
    h[row * LD + c] = (ar[c] - mean) * rstd * gm[c] + bt[c];
  if (li == 0) { stat[row * 2] = mean; stat[row * 2 + 1] = rstd; }
}

// LayerNorm backward fused with tanh backward, in place on dh
template <int C, int LD, int CPL>
static __device__ __forceinline__ void ln_tanh_bwd(float* dh, const float* a,
                                                   const float* gm, const float* stat) {
  int row = threadIdx.x >> 3, li = threadIdx.x & 7;
  float mean = stat[row * 2], rstd = stat[row * 2 + 1];
  const float* ar = a + row * LD;
  float* dr = dh + row * LD;
  float ec[CPL], xc[CPL];
  float s1 = 0.f, s2 = 0.f;
#pragma unroll
  for (int i = 0; i < CPL; ++i) {
    int c = li + 8 * i;
    float e  = dr[c] * gm[c];
    float xh = (ar[c] - mean) * rstd;
    ec[i] = e; xc[i] = xh;
    s1 += e; s2 += e * xh;
  }
  for (int m = 4; m; m >>= 1) { s1 += __shfl_xor(s1, m, 8); s2 += __shfl_xor(s2, m, 8); }
  s1 *= (1.f / C); s2 *= (1.f / C);
#pragma unroll
  for (int i = 0; i < CPL; ++i) {
    int c = li + 8 * i;
    float av = ar[c];
    float da = rstd * (ec[i] - s1 - xc[i] * s2);
    dr[c] = da * (1.f - av * av);   // tanh' = 1 - a^2
  }
}

__global__ __launch_bounds__(256) void eq_agg_kernel(
    const float* __restrict__ x,
    const float* __restrict__ W1, const float* __restrict__ b1,
    const float* __restrict__ g1, const float* __restrict__ be1,
    const float* __restrict__ W2, const float* __restrict__ b2,
    const float* __restrict__ g2, const float* __restrict__ be2,
    const float* __restrict__ W3, const float* __restrict__ b3,
    float* __restrict__ out, int bsize) {
  __shared__ float sX[TILE * LD1];    // x tile -> h1 (LN1 out) -> d1/dpre1
  __shared__ float sA1[TILE * LD1];   // tanh1 outputs (kept for backward)
  __shared__ float sA2[TILE * LD2];   // tanh2 outputs
  __shared__ float sH2[TILE * LD2];   // h2 (LN2 out) -> d2/dpre2
  __shared__ float sP[TILE * LDP];    // dp = 0.2 * p  (N=10 padded to 16)
  __shared__ float sW3[HD * LDW3];    // zero-padded W3 [64][16]
  __shared__ float sY[DIM], sYB[DIM], sS1[DIM];
  __shared__ float sB2[HD], sB3[16];
  __shared__ float sG1[DIM], sBe1[DIM], sG2[HD], sBe2[HD];
  __shared__ float sStat1[TILE * 2], sStat2[TILE * 2];

  const int tid  = threadIdx.x;
  const int wave = tid >> 5;
  const int l    = tid & 31;
  const int g    = blockIdx.x;

  // stage invariant params (W3 padded 10->16 with zeros so dp pad = 0)
  for (int i = tid; i < HD * 16; i += 256) {
    int r = i >> 4, c = i & 15;
    sW3[r * LDW3 + c] = (c < 10) ? W3[r * 10 + c] : 0.f;
  }
  if (tid < 16)  sB3[tid] = (tid < 10) ? b3[tid] : 0.f;
  if (tid < HD)  { sB2[tid] = b2[tid]; sG2[tid] = g2[tid]; sBe2[tid] = be2[tid]; }
  if (tid < DIM) { sG1[tid] = g1[tid]; sBe1[tid] = be1[tid]; sY[tid] = 0.f; }
  __syncthreads();

  for (int t = 0; t < TSTEPS; ++t) {
    // sYB = y @ W1y + b1  (z@W1 = x@W1x + y@W1y since y is identical per row)
    if (tid < DIM) {
      float acc = b1[tid];
      for (int k = 0; k < DIM; ++k)
        acc += sY[k] * W1[(DIM + k) * DIM + tid];
      sYB[tid] = acc;
      sS1[tid] = 0.f;  // column-sum accumulator of dpre1
    }
    __syncthreads();

    for (int tile = 0; tile < NODES / TILE; ++tile) {
      // ---- load x tile (coalesced b128) ----
      const float4* xs = (const float4*)(x + ((size_t)g * NODES + (size_t)tile * TILE) * DIM);
      for (int i = tid; i < TILE * (DIM / 4); i += 256) {
        int r = i >> 5, c = i & 31;
        *(float4*)(&sX[r * LD1 + c * 4]) = xs[i];
      }
      __syncthreads();

      // ---- matmul1: tanh(x@W1x + sYB) -> sA1 ; wave w owns col block w ----
      {
        v8f acc0 = {}, acc1 = {};
#pragma unroll 4
        for (int k0 = 0; k0 < DIM; k0 += 4) {
          v2f b = fragB(W1, DIM, k0, wave * 16);
          acc0 = wmma4(fragA(sX, LD1, 0, k0),  b, acc0);
          acc1 = wmma4(fragA(sX, LD1, 16, k0), b, acc1);
        }
        int col = wave * 16 + (l & 15);
        int rb  = (l >> 4) << 3;
        float yb = sYB[col];
#pragma unroll
        for (int r = 0; r < 8; ++r) {
          sA1[(rb + r) * LD1 + col]      = fast_tanh(acc0[r] + yb);
          sA1[(16 + rb + r) * LD1 + col] = fast_tanh(acc1[r] + yb);
        }
      }
      __syncthreads();

      ln_fwd<DIM, LD1>(sA1, sX, sG1, sBe1, sStat1);   // h1 overwrites x
      __syncthreads();

      // ---- matmul2: tanh(h1@W2 + b2) -> sA2 ----
      {
        int rb2 = wave >> 2, nc = (wave & 3) * 16;
        v8f acc = {};
#pragma unroll 4
        for (int k0 = 0; k0 < DIM; k0 += 4)
          acc = wmma4(fragA(sX, LD1, rb2 * 16, k0), fragB(W2, HD, k0, nc), acc);
        int col = nc + (l & 15);
        int rb  = rb2 * 16 + ((l >> 4) << 3);
        float bb = sB2[col];
#pragma unroll
        for (int r = 0; r < 8; ++r)
          sA2[(rb + r) * LD2 + col] = fast_tanh(acc[r] + bb);
      }
      __syncthreads();

      ln_fwd<HD, LD2>(sA2, sH2, sG2, sBe2, sStat2);
      __syncthreads();

      // ---- matmul3: dp = (2/10)*(h2@W3pad + b3) -> sP (waves 0,1 only) ----
      if (wave < 2) {
        v8f acc = {};
#pragma unroll
        for (int k0 = 0; k0 < HD; k0 += 4)
          acc = wmma4(fragA(sH2, LD2, wave * 16, k0), fragB(sW3, LDW3, k0, 0), acc);
        int col = l & 15;
        int rb  = wave * 16 + ((l >> 4) << 3);
        float bb = sB3[col];
#pragma unroll
        for (int r = 0; r < 8; ++r)
          sP[(rb + r) * LDP + col] = 0.2f * (acc[r] + bb);
      }
      __syncthreads();

      // ---- d2 = dp @ W3pad^T -> sH2 (h2 dead, overwrite) ----
      {
        int rb2 = wave >> 2, nc = (wave & 3) * 16;
        v8f acc = {};
#pragma unroll
        for (int k0 = 0; k0 < 16; k0 += 4)
          acc = wmma4(fragA(sP, LDP, rb2 * 16, k0), fragBT(sW3, LDW3, k0, nc), acc);
        int col = nc + (l & 15);
        int rb  = rb2 * 16 + ((l >> 4) << 3);
#pragma unroll
        for (int r = 0; r < 8; ++r)
          sH2[(rb + r) * LD2 + col] = acc[r];
      }
      __syncthreads();

      ln_tanh_bwd<HD, LD2, 8>(sH2, sA2, sG2, sStat2);  // dpre2 in place
      __syncthreads();

      // ---- d1 = dpre2 @ W2^T -> sX (h1 dead, overwrite) ----
      {
        v8f acc0 = {}, acc1 = {};
#pragma unroll 4
        for (int k0 = 0; k0 < HD; k0 += 4) {
          v2f b = fragBT(W2, HD, k0, wave * 16);
          acc0 = wmma4(fragA(sH2, LD2, 0, k0),  b, acc0);
          acc1 = wmma4(fragA(sH2, LD2, 16, k0), b, acc1);
        }
        int col = wave * 16 + (l & 15);
        int rb  = (l >> 4) << 3;
#pragma unroll
        for (int r = 0; r < 8; ++r) {
          sX[(rb + r) * LD1 + col]      = acc0[r];
          sX[(16 + rb + r) * LD1 + col] = acc1[r];
        }
      }
      __syncthreads();

      ln_tanh_bwd<DIM, LD1, 16>(sX, sA1, sG1, sStat1); // dpre1 in place
      __syncthreads();

      // ---- accumulate column sums of dpre1 (linearity: sum before @W1y^T) ----
      {
        int col = tid & 127, half = tid >> 7;
        float s = 0.f;
#pragma unroll
        for (int r = half * 16; r < half * 16 + 16; ++r) s += sX[r * LD1 + col];
        atomicAdd(&sS1[col], s);
      }
      __syncthreads();
    } // tiles

    // gy = s1 @ W1y^T + 2*REG*y/DIM ; store grads[t]; descend
    if (tid < DIM) {
      const float* w1r = W1 + (size_t)(DIM + tid) * DIM;
      float acc = 0.f;
      for (int j = 0; j < DIM; ++j) acc += sS1[j] * w1r[j];
      float gy = acc + (2.f * REGC / DIM) * sY[tid];
      out[(size_t)bsize * DIM + ((size_t)t * bsize + g) * DIM + tid] = gy;
      sY[tid] -= ALPHA * gy;
    }
    __syncthreads();
  } // steps

  if (tid < DIM) out[(size_t)g * DIM + tid] = sY[tid];
}

extern "C" void kernel_launch(void* const* d_in, const int* in_sizes, int n_in,
                              void* d_out, int out_size, void* d_ws, size_t ws_size,
                              hipStream_t stream) {
  const float* x   = (const float*)d_in[0];
  // d_in[1] = batch ids (int64) — unused: graphs are equal-size by construction
  const float* W1  = (const float*)d_in[2];
  const float* b1  = (const float*)d_in[3];
  const float* g1  = (const float*)d_in[4];
  const float* be1 = (const float*)d_in[5];
  const float* W2  = (const float*)d_in[6];
  const float* b2  = (const float*)d_in[7];
  const float* g2  = (const float*)d_in[8];
  const float* be2 = (const float*)d_in[9];
  const float* W3  = (const float*)d_in[10];
  const float* b3  = (const float*)d_in[11];
  float* out = (float*)d_out;

  const int bsize = in_sizes[0] / (NODES * DIM);  // 512
  eq_agg_kernel<<<dim3(bsize), dim3(256), 0, stream>>>(
      x, W1, b1, g1, be1, W2, b2, g2, be2, W3, b3, out, bsize);
  (void)d_ws; (void)ws_size; (void)n_in; (void)out_size;
}